// AutoencoderHom_90211493085364
// MI455X (gfx1250) — compile-verified
//
#include <hip/hip_runtime.h>
#include <hip/hip_bf16.h>
#include <math.h>

// ---------------------------------------------------------------------------
// AutoencoderHom on MI455X (gfx1250, wave32)
// f32 WMMA (V_WMMA_F32_16X16X4_F32) for all GEMMs: the homology branch needs
// ~1e-6 relative precision on pairwise distances, ruling out f16/bf16/fp8.
// GEMM: 128 threads / 4 waves -> 64 x (NSUB*16) C tile. A (row-major) and B
// (transposed) staged in LDS with stride 36 (conflict-free, 16B-aligned).
// Branch-free register prefetch: next K-tile global loads are issued before
// the WMMA block so their latency overlaps compute (no runtime guards -- the
// next-K index is a uniform branchless clamp).
// ---------------------------------------------------------------------------

typedef float v2f __attribute__((ext_vector_type(2)));
typedef float v8f __attribute__((ext_vector_type(8)));

#define BATCH   512
#define NPAIRS  130816   // 512*511/2
#define LDS_STR 36       // padded stride (floats): 16B-aligned rows, and
                         // m*36 mod 64 distinct for m=0..15 -> conflict-free

// ---------------------------------------------------------------------------
// C[M,N] = act(A[M,K] @ W[K,N] + bias); M mult of 64, K mult of 32,
// N == gridDim.x * NSUB*16 exactly. block = 128 threads (4 waves).
// ---------------------------------------------------------------------------
template <int NSUB, bool RELU>
__global__ void gemm_wmma_f32(const float* __restrict__ A,
                              const float* __restrict__ W,
                              const float* __restrict__ bias,
                              float* __restrict__ C,
                              int K, int N) {
    constexpr int TC  = NSUB * 16;    // tile cols
    constexpr int BCG = TC / 4;       // B col-groups of 4 floats
    __shared__ float As[64 * LDS_STR];
    __shared__ float Bs[TC * LDS_STR];

    const int tid   = threadIdx.x;            // 0..127
    const int wave  = tid >> 5;               // 0..3
    const int lane  = tid & 31;
    const int col0  = blockIdx.x * TC;
    const int row0  = blockIdx.y * 64;
    const int m16   = lane & 15;
    const int kh2   = (lane >> 4) * 2;        // 0 or 2

    // A cooperative-load mapping: float4, rows of 32 floats by 8 lanes
    const int a_c4 = (tid & 7) * 4;           // col 0,4,..28
    const int a_r  = tid >> 3;                // row group 0..15
    // B cooperative-load mapping: float4 across TC cols
    const int b_c4 = (tid % BCG) * 4;
    const int b_r  = tid / BCG;               // row group

    float4 ta[4], tb[NSUB];
    v8f acc[NSUB] = {};

    // prefetch first K tile into registers (clause of b128 loads)
    #pragma unroll
    for (int i = 0; i < 4; ++i)
        ta[i] = *(const float4*)&A[(size_t)(row0 + a_r + 16 * i) * K + a_c4];
    #pragma unroll
    for (int i = 0; i < NSUB; ++i)
        tb[i] = *(const float4*)&W[(size_t)(b_r + (128 / BCG) * i) * N + col0 + b_c4];

    for (int k0 = 0; k0 < K; k0 += 32) {
        __syncthreads();                      // LDS free from previous step
        // A tile 64x32 -> LDS (row-major, stride 36, 16B-aligned b128 stores)
        #pragma unroll
        for (int i = 0; i < 4; ++i)
            *(float4*)&As[(a_r + 16 * i) * LDS_STR + a_c4] = ta[i];
        // B tile 32xTC -> LDS transposed (Bs[col][k], stride 36)
        #pragma unroll
        for (int i = 0; i < NSUB; ++i) {
            const int r = b_r + (128 / BCG) * i;
            Bs[(b_c4 + 0) * LDS_STR + r] = tb[i].x;
            Bs[(b_c4 + 1) * LDS_STR + r] = tb[i].y;
            Bs[(b_c4 + 2) * LDS_STR + r] = tb[i].z;
            Bs[(b_c4 + 3) * LDS_STR + r] = tb[i].w;
        }
        __syncthreads();

        // issue next K-tile loads now; latency overlaps the WMMA block.
        // branchless clamp: last iteration redundantly reloads (in bounds).
        const int kn = (k0 + 32 < K) ? (k0 + 32) : k0;
        #pragma unroll
        for (int i = 0; i < 4; ++i)
            ta[i] = *(const float4*)&A[(size_t)(row0 + a_r + 16 * i) * K + kn + a_c4];
        #pragma unroll
        for (int i = 0; i < NSUB; ++i)
            tb[i] = *(const float4*)&W[(size_t)(kn + b_r + (128 / BCG) * i) * N + col0 + b_c4];

        // 32 K-steps of WMMA from LDS only (b64 fragment loads)
        const int arow = (wave * 16 + m16) * LDS_STR + kh2;
        #pragma unroll
        for (int kk = 0; kk < 32; kk += 4) {
            const v2f a = *(const v2f*)&As[arow + kk];
            #pragma unroll
            for (int s = 0; s < NSUB; ++s) {
                const v2f b = *(const v2f*)&Bs[(s * 16 + m16) * LDS_STR + kh2 + kk];
                acc[s] = __builtin_amdgcn_wmma_f32_16x16x4_f32(
                             false, a, false, b, (short)0, acc[s], false, false);
            }
        }
    }

    // epilogue: C/D layout -> m = r + 8*(lane>=16), n = m16
    const int mrow = row0 + wave * 16 + (kh2 >> 1) * 8;
    #pragma unroll
    for (int s = 0; s < NSUB; ++s) {
        const int cc = col0 + s * 16 + m16;
        const float bv = bias[cc];
        #pragma unroll
        for (int r = 0; r < 8; ++r) {
            float v = acc[s][r] + bv;
            if (RELU) v = fmaxf(v, 0.0f);
            C[(size_t)(mrow + r) * N + cc] = v;
        }
    }
}

// ---------------------------------------------------------------------------
// Final decoder GEMM fused with squared-error reduction against x.
// One partial per block (deterministic tree reduce, no atomics). NSUB=4.
// ---------------------------------------------------------------------------
__global__ void gemm_wmma_sse(const float* __restrict__ A,
                              const float* __restrict__ W,
                              const float* __restrict__ bias,
                              const float* __restrict__ x,
                              float* __restrict__ partials,
                              int K, int N) {
    constexpr int NSUB = 4, TC = 64, BCG = 16;
    __shared__ float As[64 * LDS_STR];
    __shared__ float Bs[TC * LDS_STR];
    __shared__ float red[128];

    const int tid   = threadIdx.x;
    const int wave  = tid >> 5;
    const int lane  = tid & 31;
    const int col0  = blockIdx.x * TC;
    const int row0  = blockIdx.y * 64;
    const int m16   = lane & 15;
    const int kh2   = (lane >> 4) * 2;

    const int a_c4 = (tid & 7) * 4;
    const int a_r  = tid >> 3;
    const int b_c4 = (tid % BCG) * 4;
    const int b_r  = tid / BCG;

    float4 ta[4], tb[NSUB];
    v8f acc[NSUB] = {};

    #pragma unroll
    for (int i = 0; i < 4; ++i)
        ta[i] = *(const float4*)&A[(size_t)(row0 + a_r + 16 * i) * K + a_c4];
    #pragma unroll
    for (int i = 0; i < NSUB; ++i)
        tb[i] = *(const float4*)&W[(size_t)(b_r + (128 / BCG) * i) * N + col0 + b_c4];

    for (int k0 = 0; k0 < K; k0 += 32) {
        __syncthreads();
        #pragma unroll
        for (int i = 0; i < 4; ++i)
            *(float4*)&As[(a_r + 16 * i) * LDS_STR + a_c4] = ta[i];
        #pragma unroll
        for (int i = 0; i < NSUB; ++i) {
            const int r = b_r + (128 / BCG) * i;
            Bs[(b_c4 + 0) * LDS_STR + r] = tb[i].x;
            Bs[(b_c4 + 1) * LDS_STR + r] = tb[i].y;
            Bs[(b_c4 + 2) * LDS_STR + r] = tb[i].z;
            Bs[(b_c4 + 3) * LDS_STR + r] = tb[i].w;
        }
        __syncthreads();

        const int kn = (k0 + 32 < K) ? (k0 + 32) : k0;
        #pragma unroll
        for (int i = 0; i < 4; ++i)
            ta[i] = *(const float4*)&A[(size_t)(row0 + a_r + 16 * i) * K + kn + a_c4];
        #pragma unroll
        for (int i = 0; i < NSUB; ++i)
            tb[i] = *(const float4*)&W[(size_t)(kn + b_r + (128 / BCG) * i) * N + col0 + b_c4];

        const int arow = (wave * 16 + m16) * LDS_STR + kh2;
        #pragma unroll
        for (int kk = 0; kk < 32; kk += 4) {
            const v2f a = *(const v2f*)&As[arow + kk];
            #pragma unroll
            for (int s = 0; s < NSUB; ++s) {
                const v2f b = *(const v2f*)&Bs[(s * 16 + m16) * LDS_STR + kh2 + kk];
                acc[s] = __builtin_amdgcn_wmma_f32_16x16x4_f32(
                             false, a, false, b, (short)0, acc[s], false, false);
            }
        }
    }

    const int mrow = row0 + wave * 16 + (kh2 >> 1) * 8;
    float local = 0.0f;
    #pragma unroll
    for (int s = 0; s < NSUB; ++s) {
        const int cc = col0 + s * 16 + m16;
        const float bv = bias[cc];
        #pragma unroll
        for (int r = 0; r < 8; ++r) {
            const float v = acc[s][r] + bv;
            const float d = x[(size_t)(mrow + r) * N + cc] - v;
            local += d * d;
        }
    }
    red[tid] = local;
    __syncthreads();
    for (int s = 64; s; s >>= 1) {
        if (tid < s) red[tid] += red[tid + s];
        __syncthreads();
    }
    if (tid == 0) partials[blockIdx.y * gridDim.x + blockIdx.x] = red[0];
}

// ---------------------------------------------------------------------------
// Per-column mean / std (ddof=1) of latent, compactness partial per column,
// and standardized latent LH. One block per column (32 blocks, 256 threads).
// ---------------------------------------------------------------------------
__global__ void colstats_kernel(const float* __restrict__ lat,
                                float* __restrict__ comp,
                                float* __restrict__ lh) {
    const int c = blockIdx.x;      // 0..31
    const int t = threadIdx.x;     // 0..255
    __shared__ float red[256];
    __shared__ float mu_s, sd_s;

    float s = 0.0f;
    for (int i = t; i < BATCH; i += 256) s += lat[i * 32 + c];
    red[t] = s; __syncthreads();
    for (int k = 128; k; k >>= 1) { if (t < k) red[t] += red[t + k]; __syncthreads(); }
    if (t == 0) mu_s = red[0] * (1.0f / (float)BATCH);
    __syncthreads();
    const float mu = mu_s;

    float v = 0.0f, a = 0.0f;
    for (int i = t; i < BATCH; i += 256) {
        float d = lat[i * 32 + c] - mu;
        v += d * d;
        a += fabsf(d);
    }
    red[t] = v; __syncthreads();
    for (int k = 128; k; k >>= 1) { if (t < k) red[t] += red[t + k]; __syncthreads(); }
    if (t == 0) sd_s = sqrtf(red[0] * (1.0f / (float)(BATCH - 1)));
    __syncthreads();
    const float inv = 1.0f / sd_s;

    red[t] = a; __syncthreads();
    for (int k = 128; k; k >>= 1) { if (t < k) red[t] += red[t + k]; __syncthreads(); }
    if (t == 0) comp[c] = red[0];

    for (int i = t; i < BATCH; i += 256)
        lh[i * 32 + c] = (lat[i * 32 + c] - mu) * inv;
}

// ---------------------------------------------------------------------------
// Pairwise distances + death matching. 16x16 pair tile per block.
// pd/flag written at upper-triangular flat index k (np.triu_indices order).
// ---------------------------------------------------------------------------
__global__ void pdist_flags_kernel(const float* __restrict__ lh,
                                   const float* __restrict__ deaths, int nd,
                                   float* __restrict__ pd_arr,
                                   unsigned char* __restrict__ flag_arr) {
    const int bi = blockIdx.y, bj = blockIdx.x;
    if (bj < bi) return;   // entirely below diagonal (uniform exit)
    __shared__ float li[16][32];
    __shared__ float lj[16][32];
    __shared__ float dlo[512], dhi[512];
    const int t = threadIdx.x;   // 0..255

    for (int j = t; j < nd; j += 256) {
        float d   = deaths[j];
        float tol = 1e-8f + 1e-6f * fabsf(d);
        dlo[j] = d - tol;
        dhi[j] = d + tol;
    }
    for (int e = t; e < 512; e += 256) {
        int r = e >> 5, c = e & 31;
        li[r][c] = lh[(bi * 16 + r) * 32 + c];
        lj[r][c] = lh[(bj * 16 + r) * 32 + c];
    }
    __syncthreads();

    const int ti = t >> 4, tj = t & 15;
    const int i = bi * 16 + ti, j = bj * 16 + tj;
    if (j <= i) return;
    float d2 = 0.0f;
    #pragma unroll
    for (int c = 0; c < 32; ++c) {
        float d = li[ti][c] - lj[tj][c];
        d2 += d * d;
    }
    const float p = sqrtf(d2);
    unsigned char f = 0;
    for (int jj = 0; jj < nd; ++jj) {
        if (p >= dlo[jj] && p <= dhi[jj]) { f = 1; break; }
    }
    const int k = i * (2 * BATCH - i - 1) / 2 + (j - i - 1);
    pd_arr[k]   = p;
    flag_arr[k] = f;
}

// ---------------------------------------------------------------------------
// Ordered prefix selection: sum pd over the first `nsel` flagged pairs in
// flat-index order. Single block, 1024 threads, chunked Hillis-Steele scan.
// ---------------------------------------------------------------------------
__global__ void hom_select_kernel(const float* __restrict__ pd,
                                  const unsigned char* __restrict__ flag,
                                  int npairs, int nsel,
                                  float* __restrict__ hom_out) {
    __shared__ int   scan[1024];
    __shared__ float redf[1024];
    __shared__ int   base_s;
    const int t = threadIdx.x;
    if (t == 0) base_s = 0;
    float acc = 0.0f;
    __syncthreads();

    for (int start = 0; start < npairs; start += 1024) {
        const int idx = start + t;
        const int f = (idx < npairs) ? (int)flag[idx] : 0;
        const float p = f ? pd[idx] : 0.0f;
        scan[t] = f; __syncthreads();
        for (int off = 1; off < 1024; off <<= 1) {
            int v = (t >= off) ? scan[t - off] : 0;
            __syncthreads();
            scan[t] += v;
            __syncthreads();
        }
        const int excl = scan[t] - f;
        const int base = base_s;
        if (f && (base + excl) < nsel) acc += fabsf(p);
        __syncthreads();
        if (t == 1023) base_s = base + scan[1023];
        __syncthreads();
    }

    redf[t] = acc; __syncthreads();
    for (int s = 512; s; s >>= 1) { if (t < s) redf[t] += redf[t + s]; __syncthreads(); }
    if (t == 0) hom_out[0] = redf[0];
}

// ---------------------------------------------------------------------------
// Final combine: loss = mse + 0.1*hom + 0.01*comp   (one block, 512 threads)
// ---------------------------------------------------------------------------
__global__ void finalize_kernel(const float* __restrict__ sse_part,
                                const float* __restrict__ comp,
                                const float* __restrict__ hom,
                                float* __restrict__ out, float inv_count) {
    __shared__ float red[512];
    __shared__ float sse_s;
    const int t = threadIdx.x;
    red[t] = sse_part[t]; __syncthreads();
    for (int s = 256; s; s >>= 1) { if (t < s) red[t] += red[t + s]; __syncthreads(); }
    if (t == 0) sse_s = red[0];
    __syncthreads();
    red[t] = (t < 32) ? comp[t] : 0.0f; __syncthreads();
    for (int s = 256; s; s >>= 1) { if (t < s) red[t] += red[t + s]; __syncthreads(); }
    if (t == 0) out[0] = sse_s * inv_count + 0.1f * hom[0] + 0.01f * red[0];
}

// ---------------------------------------------------------------------------
// Launcher
// ---------------------------------------------------------------------------
extern "C" void kernel_launch(void* const* d_in, const int* in_sizes, int n_in,
                              void* d_out, int out_size, void* d_ws, size_t ws_size,
                              hipStream_t stream) {
    const float* x      = (const float*)d_in[0];   // 512x1024
    // d_in[1] births (512, all zero; only its size matters)
    const float* deaths = (const float*)d_in[2];   // 511
    const float* We0 = (const float*)d_in[3];
    const float* be0 = (const float*)d_in[4];
    const float* We1 = (const float*)d_in[5];
    const float* be1 = (const float*)d_in[6];
    const float* We2 = (const float*)d_in[7];
    const float* be2 = (const float*)d_in[8];
    const float* Wd0 = (const float*)d_in[9];
    const float* bd0 = (const float*)d_in[10];
    const float* Wd1 = (const float*)d_in[11];
    const float* bd1 = (const float*)d_in[12];
    const float* Wd2 = (const float*)d_in[13];
    const float* bd2 = (const float*)d_in[14];

    const int nd   = in_sizes[2];                       // 511 deaths
    const int nsel = (in_sizes[1] < in_sizes[2]) ? in_sizes[1] : in_sizes[2];

    // workspace carve-up (floats). D1/D2 reuse H1/H2.
    float* ws  = (float*)d_ws;
    float* H1  = ws;                  // 512*512
    float* H2  = H1 + 512 * 512;      // 512*512
    float* LAT = H2 + 512 * 512;      // 512*32
    float* LH  = LAT + 512 * 32;      // 512*32
    float* COMP = LH + 512 * 32;      // 32
    float* HOM  = COMP + 32;          // 1 (padded to 32)
    float* PD   = HOM + 32;           // NPAIRS
    float* SSEP = PD + NPAIRS;        // 512 partials
    unsigned char* FLAG = (unsigned char*)(SSEP + 512); // NPAIRS bytes
    float* D1 = H1;
    float* D2 = H2;

    const dim3 blk(128);

    // Encoder (grid: N/(NSUB*16) x M/64)
    gemm_wmma_f32<4, true ><<<dim3(8, 8), blk, 0, stream>>>(x,  We0, be0, H1, 1024, 512);
    gemm_wmma_f32<4, true ><<<dim3(8, 8), blk, 0, stream>>>(H1, We1, be1, H2, 512,  512);
    gemm_wmma_f32<2, false><<<dim3(1, 8), blk, 0, stream>>>(H2, We2, be2, LAT, 512, 32);

    // Standardize + compactness partials
    colstats_kernel<<<dim3(32), dim3(256), 0, stream>>>(LAT, COMP, LH);

    // Pairwise distances + death matching
    pdist_flags_kernel<<<dim3(32, 32), dim3(256), 0, stream>>>(LH, deaths, nd, PD, FLAG);

    // Ordered selection of first nsel flagged pairs
    hom_select_kernel<<<dim3(1), dim3(1024), 0, stream>>>(PD, FLAG, NPAIRS, nsel, HOM);

    // Decoder (D1/D2 reuse H1/H2)
    gemm_wmma_f32<4, true ><<<dim3(8, 8),  blk, 0, stream>>>(LAT, Wd0, bd0, D1, 32,  512);
    gemm_wmma_f32<4, true ><<<dim3(8, 8),  blk, 0, stream>>>(D1,  Wd1, bd1, D2, 512, 512);
    gemm_wmma_sse<<<dim3(16, 8), blk, 0, stream>>>(D2, Wd2, bd2, x, SSEP, 512, 1024);

    // Combine
    finalize_kernel<<<dim3(1), dim3(512), 0, stream>>>(SSEP, COMP, HOM, (float*)d_out,
                                                       1.0f / (512.0f * 1024.0f));
}